// GCN_25769803994
// MI455X (gfx1250) — compile-verified
//
#include <hip/hip_runtime.h>

typedef float v2f __attribute__((ext_vector_type(2)));
typedef float v8f __attribute__((ext_vector_type(8)));

#define NEG_SLOPE 0.2f
#define ENC_NEG_INF 0x007FFFFFu  // ordered-uint encoding of -inf

__device__ __forceinline__ unsigned f32_ord_enc(float f) {
  unsigned u = __float_as_uint(f);
  return (u & 0x80000000u) ? ~u : (u | 0x80000000u);
}
__device__ __forceinline__ float f32_ord_dec(unsigned u) {
  return __uint_as_float((u & 0x80000000u) ? (u & 0x7FFFFFFFu) : ~u);
}

// -------------------- fills --------------------
__global__ void k_fill_f32(float* __restrict__ p, float v, int n) {
  int i = blockIdx.x * blockDim.x + threadIdx.x;
  if (i < n) p[i] = v;
}
__global__ void k_fill_u32(unsigned* __restrict__ p, unsigned v, int n) {
  int i = blockIdx.x * blockDim.x + threadIdx.x;
  if (i < n) p[i] = v;
}

// -------------------- layer-1 GEMM via f32 WMMA --------------------
// feat1[N,128] = X[N,128] @ W1[128,128]
// One wave computes one 16x16 output tile with V_WMMA_F32_16X16X4_F32,
// fully unrolled K-loop of 32. 8 waves/block cover the 8 column tiles of
// one 16-row strip.
__global__ void k_gemm1_wmma(const float* __restrict__ X,
                             const float* __restrict__ W,
                             float* __restrict__ F, int N) {
  const int wave = threadIdx.x >> 5;     // 0..7 -> column tile
  const int lane = threadIdx.x & 31;
  const int lo   = lane & 15;
  const int hi   = lane >> 4;
  const int rbase = blockIdx.x * 16;
  int arow = rbase + lo;
  if (arow >= N) arow = N - 1;           // clamp: keep EXEC all-ones for WMMA
  const int n0 = wave * 16;

  const float* __restrict__ xrow = X + (size_t)arow * 128;
  v8f c = {};
  #pragma unroll
  for (int k0 = 0; k0 < 128; k0 += 4) {
    const int ka = k0 + 2 * hi;          // A: lanes<16 K=k0..k0+1, lanes>=16 K=k0+2..k0+3
    v2f a;
    a.x = xrow[ka];
    a.y = xrow[ka + 1];
    v2f b;                               // B mirrors A with N across lanes
    b.x = W[(size_t)ka * 128 + n0 + lo];
    b.y = W[(size_t)(ka + 1) * 128 + n0 + lo];
    c = __builtin_amdgcn_wmma_f32_16x16x4_f32(false, a, false, b,
                                              (short)0, c, false, false);
  }
  if (rbase + 16 <= N) {                 // uniform fast path: no per-store EXEC games
    float* __restrict__ frow = F + (size_t)(rbase + 8 * hi) * 128 + n0 + lo;
    #pragma unroll
    for (int i = 0; i < 8; ++i) frow[(size_t)i * 128] = c[i];
  } else {
    #pragma unroll
    for (int i = 0; i < 8; ++i) {        // tail strip (unused for N%16==0)
      const int orow = rbase + i + 8 * hi;
      if (orow < N) F[(size_t)orow * 128 + n0 + lo] = c[i];
    }
  }
}

// -------------------- layer-1 attention scores --------------------
// el1[n,h] = dot(feat1[n,h,:], al1[h,:]) ; er1 likewise. H=2, D=64.
__global__ void k_scores1(const float* __restrict__ feat1,
                          const float* __restrict__ al1,
                          const float* __restrict__ ar1,
                          float* __restrict__ el1, float* __restrict__ er1,
                          int N) {
  int i = blockIdx.x * blockDim.x + threadIdx.x;
  if (i >= N * 2) return;
  const int n = i >> 1, h = i & 1;
  const float* fp = feat1 + (size_t)n * 128 + h * 64;
  const float* ap = al1 + h * 64;
  const float* rp = ar1 + h * 64;
  float sl = 0.f, sr = 0.f;
  #pragma unroll
  for (int d = 0; d < 64; d += 4) {
    float4 f = *(const float4*)(fp + d);
    float4 a = *(const float4*)(ap + d);
    float4 r = *(const float4*)(rp + d);
    sl += f.x * a.x + f.y * a.y + f.z * a.z + f.w * a.w;
    sr += f.x * r.x + f.y * r.y + f.z * r.z + f.w * r.w;
  }
  el1[i] = sl;
  er1[i] = sr;
}

// -------------------- edge softmax passes (generic in H) --------------------
template <int H>
__global__ void k_edge_score_max(const float* __restrict__ el,
                                 const float* __restrict__ er,
                                 const int* __restrict__ src,
                                 const int* __restrict__ dst,
                                 float* __restrict__ ebuf,
                                 unsigned* __restrict__ menc, int E) {
  int i = blockIdx.x * blockDim.x + threadIdx.x;
  if (i >= E * H) return;
  const int e = i / H, h = i % H;
  const int s = src[e], d = dst[e];
  float v = el[s * H + h] + er[d * H + h];
  v = (v >= 0.f) ? v : NEG_SLOPE * v;    // leaky_relu
  ebuf[i] = v;
  atomicMax(&menc[d * H + h], f32_ord_enc(v));
}

template <int H>
__global__ void k_edge_expsum(float* __restrict__ ebuf,
                              const unsigned* __restrict__ menc,
                              const int* __restrict__ dst,
                              float* __restrict__ denom, int E) {
  int i = blockIdx.x * blockDim.x + threadIdx.x;
  if (i >= E * H) return;
  const int e = i / H, h = i % H;
  const int d = dst[e];
  float m = f32_ord_dec(menc[d * H + h]);
  if (!(m >= -3.0e38f && m <= 3.0e38f)) m = 0.f;  // finite guard (matches ref)
  const float ex = expf(ebuf[i] - m);
  ebuf[i] = ex;                                    // overwrite with exp value
  atomicAdd(&denom[d * H + h], ex);
}

// -------------------- layer-1 aggregation (H=2, D=64 -> 128 f32 per node) ----
// 32 lanes per edge, float4 per lane.
__global__ void k_edge_aggr1(const float* __restrict__ ebuf,
                             const float* __restrict__ denom,
                             const float* __restrict__ feat,
                             const int* __restrict__ src,
                             const int* __restrict__ dst,
                             float* __restrict__ out, int E) {
  int t = blockIdx.x * blockDim.x + threadIdx.x;
  const int e = t >> 5;
  if (e >= E) return;
  const int c = (t & 31) * 4;            // feature column 0..127
  const int h = c >> 6;                  // head
  const int s = src[e], d = dst[e];
  const float alpha = ebuf[e * 2 + h] / fmaxf(denom[d * 2 + h], 1e-30f);
  const float4 f = *(const float4*)(feat + (size_t)s * 128 + c);
  float* o = out + (size_t)d * 128 + c;
  atomicAdd(o + 0, alpha * f.x);
  atomicAdd(o + 1, alpha * f.y);
  atomicAdd(o + 2, alpha * f.z);
  atomicAdd(o + 3, alpha * f.w);
}

// -------------------- head-mean + bias + relu -> h[N,64] --------------------
__global__ void k_headmean_relu(const float* __restrict__ out1,
                                const float* __restrict__ b1,
                                float* __restrict__ hbuf, int N) {
  int i = blockIdx.x * blockDim.x + threadIdx.x;
  if (i >= N * 64) return;
  const int n = i >> 6, d = i & 63;
  float v = (out1[(size_t)n * 128 + d] + b1[d] +
             out1[(size_t)n * 128 + 64 + d] + b1[64 + d]) * 0.5f;
  hbuf[i] = v > 0.f ? v : 0.f;
}

// -------------------- layer-2 projection + scores (tiny: 64x2) --------------
__global__ void k_feat2_scores(const float* __restrict__ hbuf,
                               const float* __restrict__ W2,
                               const float* __restrict__ al2,
                               const float* __restrict__ ar2,
                               float* __restrict__ feat2,
                               float* __restrict__ el2,
                               float* __restrict__ er2, int N) {
  __shared__ float sW[128];
  if (threadIdx.x < 128) sW[threadIdx.x] = W2[threadIdx.x];
  __syncthreads();
  int n = blockIdx.x * blockDim.x + threadIdx.x;
  if (n >= N) return;
  const float* hp = hbuf + (size_t)n * 64;
  float f0 = 0.f, f1 = 0.f;
  #pragma unroll 8
  for (int k = 0; k < 64; ++k) {
    const float hv = hp[k];
    f0 += hv * sW[k * 2 + 0];
    f1 += hv * sW[k * 2 + 1];
  }
  feat2[n * 2 + 0] = f0;
  feat2[n * 2 + 1] = f1;
  el2[n] = f0 * al2[0] + f1 * al2[1];
  er2[n] = f0 * ar2[0] + f1 * ar2[1];
}

// -------------------- layer-2 aggregation (H=1, D=2) ------------------------
__global__ void k_edge_aggr2(const float* __restrict__ ebuf,
                             const float* __restrict__ denom,
                             const float* __restrict__ feat2,
                             const int* __restrict__ src,
                             const int* __restrict__ dst,
                             float* __restrict__ out, int E) {
  int e = blockIdx.x * blockDim.x + threadIdx.x;
  if (e >= E) return;
  const int s = src[e], d = dst[e];
  const float alpha = ebuf[e] / fmaxf(denom[d], 1e-30f);
  atomicAdd(out + (size_t)d * 2 + 0, alpha * feat2[s * 2 + 0]);
  atomicAdd(out + (size_t)d * 2 + 1, alpha * feat2[s * 2 + 1]);
}

__global__ void k_bias2(float* __restrict__ out, const float* __restrict__ b2,
                        int N) {
  int i = blockIdx.x * blockDim.x + threadIdx.x;
  if (i >= N * 2) return;
  out[i] += b2[i & 1];
}

// -------------------- launcher --------------------
static inline int cdiv(long long a, long long b) { return (int)((a + b - 1) / b); }

extern "C" void kernel_launch(void* const* d_in, const int* in_sizes, int n_in,
                              void* d_out, int out_size, void* d_ws, size_t ws_size,
                              hipStream_t stream) {
  const float* x   = (const float*)d_in[0];
  const int*   src = (const int*)d_in[1];
  const int*   dst = (const int*)d_in[2];
  const float* W1  = (const float*)d_in[3];
  const float* al1 = (const float*)d_in[4];
  const float* ar1 = (const float*)d_in[5];
  const float* b1  = (const float*)d_in[6];
  const float* W2  = (const float*)d_in[7];
  const float* al2 = (const float*)d_in[8];
  const float* ar2 = (const float*)d_in[9];
  const float* b2  = (const float*)d_in[10];

  const int N = in_sizes[0] / 128;   // 100000
  const int E = in_sizes[1];         // 1600000

  float* ws = (float*)d_ws;
  size_t off = 0;
  float*    feat1 = ws + off; off += (size_t)N * 128;
  float*    out1  = ws + off; off += (size_t)N * 128;
  float*    ebuf  = ws + off; off += (size_t)E * 2;   // layer2 reuses (E*1)
  float*    hbuf  = ws + off; off += (size_t)N * 64;
  float*    el1   = ws + off; off += (size_t)N * 2;
  float*    er1   = ws + off; off += (size_t)N * 2;
  unsigned* m1    = (unsigned*)(ws + off); off += (size_t)N * 2;
  float*    den1  = ws + off; off += (size_t)N * 2;
  float*    feat2 = ws + off; off += (size_t)N * 2;
  float*    el2   = ws + off; off += (size_t)N;
  float*    er2   = ws + off; off += (size_t)N;
  unsigned* m2    = (unsigned*)(ws + off); off += (size_t)N;
  float*    den2  = ws + off; off += (size_t)N;
  float*    outf  = (float*)d_out;

  const int B = 256;

  // init accumulators (harness does not re-zero between replays)
  k_fill_f32<<<cdiv((long long)N * 128, B), B, 0, stream>>>(out1, 0.f, N * 128);
  k_fill_u32<<<cdiv(N * 2, B), B, 0, stream>>>(m1, ENC_NEG_INF, N * 2);
  k_fill_f32<<<cdiv(N * 2, B), B, 0, stream>>>(den1, 0.f, N * 2);
  k_fill_u32<<<cdiv(N, B), B, 0, stream>>>(m2, ENC_NEG_INF, N);
  k_fill_f32<<<cdiv(N, B), B, 0, stream>>>(den2, 0.f, N);
  k_fill_f32<<<cdiv(N * 2, B), B, 0, stream>>>(outf, 0.f, N * 2);

  // ---- layer 1 ----
  k_gemm1_wmma<<<cdiv(N, 16), 256, 0, stream>>>(x, W1, feat1, N);
  k_scores1<<<cdiv(N * 2, B), B, 0, stream>>>(feat1, al1, ar1, el1, er1, N);
  k_edge_score_max<2><<<cdiv((long long)E * 2, B), B, 0, stream>>>(
      el1, er1, src, dst, ebuf, m1, E);
  k_edge_expsum<2><<<cdiv((long long)E * 2, B), B, 0, stream>>>(
      ebuf, m1, dst, den1, E);
  k_edge_aggr1<<<cdiv((long long)E * 32, B), B, 0, stream>>>(
      ebuf, den1, feat1, src, dst, out1, E);
  k_headmean_relu<<<cdiv((long long)N * 64, B), B, 0, stream>>>(out1, b1, hbuf, N);

  // ---- layer 2 ----
  k_feat2_scores<<<cdiv(N, B), B, 0, stream>>>(hbuf, W2, al2, ar2,
                                               feat2, el2, er2, N);
  k_edge_score_max<1><<<cdiv(E, B), B, 0, stream>>>(el2, er2, src, dst, ebuf, m2, E);
  k_edge_expsum<1><<<cdiv(E, B), B, 0, stream>>>(ebuf, m2, dst, den2, E);
  k_edge_aggr2<<<cdiv(E, B), B, 0, stream>>>(ebuf, den2, feat2, src, dst, outf, E);
  k_bias2<<<cdiv(N * 2, B), B, 0, stream>>>(outf, b2, N);
}